// Loss_10952166604854
// MI455X (gfx1250) — compile-verified
//
#include <hip/hip_runtime.h>
#include <math.h>
#include <float.h>

// ---------------------------------------------------------------------------
// CenterNet-style loss (weighted-Hausdorff heatmap + bounded-IoU) for MI455X.
//
// Heavy part: per-(s,b) distance matrix d(i,k) between 16384 grid cells and
// 128 points. Factorized as rank-4 product so each 16x16 tile of d^2 is ONE
// v_wmma_f32_16x16x4_f32:
//   A row i  = [y, x, y^2+x^2, 1]
//   B col k  = [-2*py, -2*px, 1, py^2+px^2]
// VALU/TRANS then do clamp -> v_sqrt -> masked-min (term1) and v_rcp +
// 3 squarings for w^-9 soft-min accumulation (term2), co-executing with the
// matrix pipe. __launch_bounds__(256,1): only 32 blocks exist, so trade
// occupancy for VGPRs and keep the 8 B-fragments + pipelined C tiles
// spill-free in the inner loop.
// ---------------------------------------------------------------------------

typedef __attribute__((ext_vector_type(2))) float v2f;
typedef __attribute__((ext_vector_type(8))) float v8f;

#define EPSF     1e-6f
#define BETA     0.2f
#define SS       2
#define BB       16
#define KK       128
#define HH       128
#define WWID     128
#define HWN      (HH * WWID)
#define NTILES   (HWN / 16)      // 1024 row tiles of 16 grid points

__device__ __forceinline__ float wave_sum32(float v) {
  #pragma unroll
  for (int o = 16; o; o >>= 1) v += __shfl_xor(v, o, 32);
  return v;
}

__device__ __forceinline__ float smooth_l1(float l) {
  return (l < BETA) ? (0.5f * l * l / BETA) : (l - 0.5f * BETA);
}

// Raw hardware ops: single v_sqrt_f32 / v_rcp_f32 / v_med3_f32.
__device__ __forceinline__ float hw_sqrt(float x) { return __builtin_amdgcn_sqrtf(x); }
__device__ __forceinline__ float hw_rcp(float x)  { return __builtin_amdgcn_rcpf(x); }
__device__ __forceinline__ float clamp_pos(float x) {
  return __builtin_amdgcn_fmed3f(x, 0.0f, FLT_MAX);   // finite hi keeps v_med3
}

// ---------------------------------------------------------------------------
// Kernel 1: weighted Hausdorff distance term per (s,b). 32 blocks x 256 thr.
// Dynamic LDS: p[HWN] + accK[KK] + red[8]  (~66 KB; WGP has 320 KB).
// ---------------------------------------------------------------------------
__global__ void __launch_bounds__(256, 1)
whd_kernel(const float* __restrict__ hm,
           const int* __restrict__ ctr,
           const int* __restrict__ reg_mask,
           float* __restrict__ ws) {
  extern __shared__ float smem[];
  float* p_lds = smem;             // HWN floats: clipped sigmoid heatmap
  float* accK  = smem + HWN;       // KK floats : sum_i (w+eps)^-9 per point
  float* red   = accK + KK;        // [0]=n_est, [1]=term1 numerator

  const int blk  = blockIdx.x;     // blk = s*BB + b
  const int b    = blk % BB;
  const int tid  = threadIdx.x;
  const int lane = tid & 31;
  const int wave = tid >> 5;
  const int colLane = lane & 15;
  const float max_dist = hw_sqrt((float)(HH * HH + WWID * WWID));

  if (tid < KK) accK[tid] = 0.0f;
  if (tid == 0) { red[0] = 0.0f; red[1] = 0.0f; }
  __syncthreads();

  // ---- stage p = clip(sigmoid(hm), 1e-4, 1-1e-4) into LDS; accumulate n_est
  const float* hmb = hm + (size_t)blk * HWN;
  float nPart = 0.0f;
  for (int i = tid; i < HWN; i += 256) {
    float v = hmb[i];
    float p = hw_rcp(1.0f + __expf(-v));
    p = __builtin_amdgcn_fmed3f(p, 1e-4f, 1.0f - 1e-4f);
    p_lds[i] = p;
    nPart += p;
  }
  nPart = wave_sum32(nPart);
  if (lane == 0) atomicAdd(&red[0], nPart);

  // ---- per-wave B fragments (8 groups of 16 points) + column masks
  // B 4x16 layout: lanes 0-15 carry K=0,1; lanes 16-31 carry K=2,3; N=lane&15
  v2f   bfrag[8];
  float mcol[8];
#if !__has_builtin(__builtin_amdgcn_wmma_f32_16x16x4_f32)
  float pyv[8], pxv[8];
#endif
  #pragma unroll
  for (int g = 0; g < 8; ++g) {
    int k = g * 16 + colLane;
    // pts = ctr[..., ::-1] -> y from ctr[...,1], x from ctr[...,0]
    float py = (float)ctr[(b * KK + k) * 2 + 1];
    float px = (float)ctr[(b * KK + k) * 2 + 0];
#if !__has_builtin(__builtin_amdgcn_wmma_f32_16x16x4_f32)
    pyv[g] = py; pxv[g] = px;
#endif
    if (lane < 16) { bfrag[g][0] = -2.0f * py; bfrag[g][1] = -2.0f * px; }
    else           { bfrag[g][0] = 1.0f;       bfrag[g][1] = py * py + px * px; }
    mcol[g] = (float)reg_mask[b * KK + k];
  }

  float powacc[8];
  #pragma unroll
  for (int g = 0; g < 8; ++g) powacc[g] = 0.0f;
  float t1 = 0.0f;

  __syncthreads();   // p_lds ready

  // ---- main loop: each wave walks 16-row grid tiles
  for (int t = wave; t < NTILES; t += 8) {
    const int tb = t * 16;

    // A 16x4 layout: M=lane&15; lanes 0-15 carry K=0,1; lanes 16-31 K=2,3
    int ia = tb + colLane;
    float ya = (float)(ia >> 7);          // W == 128
    float xa = (float)(ia & (WWID - 1));
    v2f a;
    if (lane < 16) { a[0] = ya;            a[1] = xa;   }
    else           { a[0] = ya*ya + xa*xa; a[1] = 1.0f; }

    // p for the C-layout rows: reg r holds row M = r + 8*(lane>=16)
    const int rbase = tb + ((lane >> 4) << 3);
    float pf[8];
    #pragma unroll
    for (int r = 0; r < 8; ++r) pf[r] = p_lds[rbase + r];   // -> ds_load_b128 x2

    float dmin[8];
    #pragma unroll
    for (int r = 0; r < 8; ++r) dmin[r] = max_dist;

    #pragma unroll
    for (int g = 0; g < 8; ++g) {
#if __has_builtin(__builtin_amdgcn_wmma_f32_16x16x4_f32)
      v8f c = {0.f, 0.f, 0.f, 0.f, 0.f, 0.f, 0.f, 0.f};
      c = __builtin_amdgcn_wmma_f32_16x16x4_f32(
              /*neg_a=*/false, a, /*neg_b=*/false, bfrag[g],
              /*c_mod=*/(short)0, c, /*reuse_a=*/false, /*reuse_b=*/false);
#else
#warning "wmma_f32_16x16x4_f32 builtin unavailable -- VALU fallback in use"
      v8f c;
      #pragma unroll
      for (int r = 0; r < 8; ++r) {
        int row = rbase + r;
        float dy = (float)(row >> 7) - pyv[g];
        float dx = (float)(row & (WWID - 1)) - pxv[g];
        c[r] = dy * dy + dx * dx;
      }
      (void)a;
#endif
      const float mc = mcol[g];
      #pragma unroll
      for (int r = 0; r < 8; ++r) {
        float d = hw_sqrt(clamp_pos(c[r]));            // v_med3 + v_sqrt
        float cand = (mc > 0.5f) ? d : max_dist;       // invalid pts excluded
        dmin[r] = fminf(dmin[r], cand);
        // weighted = (1-p)*max_dist + p*d = max_dist + p*(d - max_dist)
        float w = fmaf(pf[r], d - max_dist, max_dist) + EPSF;
        float u  = hw_rcp(w);                          // w^-9 via squarings
        float u2 = u * u, u4 = u2 * u2, u8 = u4 * u4;
        powacc[g] = fmaf(u8, u, powacc[g]);
      }
    }

    // min over the 16 columns within each lane-half (xor<16 stays in half)
    #pragma unroll
    for (int r = 0; r < 8; ++r) {
      float dm = dmin[r];
      dm = fminf(dm, __shfl_xor(dm, 8, 32));
      dm = fminf(dm, __shfl_xor(dm, 4, 32));
      dm = fminf(dm, __shfl_xor(dm, 2, 32));
      dm = fminf(dm, __shfl_xor(dm, 1, 32));
      if (colLane == 0) t1 = fmaf(pf[r], dm, t1);      // term1 numerator
    }
  }

  // ---- fold lane halves (both halves accumulated the same 16 columns)
  #pragma unroll
  for (int g = 0; g < 8; ++g) {
    float v = powacc[g] + __shfl_xor(powacc[g], 16, 32);
    if (lane < 16) atomicAdd(&accK[g * 16 + lane], v); // ds_add_f32
  }
  t1 = wave_sum32(t1);
  if (lane == 0) atomicAdd(&red[1], t1);
  __syncthreads();

  if (tid == 0) {
    float n_est = red[0];
    float term1 = red[1] / (n_est + EPSF);
    float msum = 0.0f, t2num = 0.0f;
    for (int k = 0; k < KK; ++k) {
      float mk   = (float)reg_mask[b * KK + k];
      float mean = accK[k] * (1.0f / (float)HWN);
      float minn = __powf(mean, -1.0f / 9.0f);         // mean^(1/P), P=-9
      t2num = fmaf(minn, mk, t2num);
      msum += mk;
    }
    float term2 = t2num / fmaxf(msum, 1.0f);
    float valid = (msum > 0.0f) ? 1.0f : 0.0f;
    ws[blk] = (term1 + term2) * valid / (float)(SS * BB);
  }
}

// ---------------------------------------------------------------------------
// Kernel 2: bounded-IoU smooth-L1 per b. 16 blocks x 128 threads (k = tid).
// ---------------------------------------------------------------------------
__global__ void iou_kernel(const float* __restrict__ wh_map,
                           const float* __restrict__ reg_map,
                           const float* __restrict__ reg_gt,
                           const float* __restrict__ wh_gt,
                           const int* __restrict__ ind,
                           const int* __restrict__ reg_mask,
                           float* __restrict__ ws) {
  __shared__ float sr[4][3];
  const int b    = blockIdx.x;
  const int k    = threadIdx.x;
  const int lane = k & 31;
  const int wave = k >> 5;

  const float mf  = (float)reg_mask[b * KK + k];
  const int   iv  = ind[b * KK + k];
  const float rg0 = reg_gt[(b * KK + k) * 2 + 0];
  const float rg1 = reg_gt[(b * KK + k) * 2 + 1];
  const float wg  = wh_gt[(b * KK + k) * 2 + 0];
  const float hg  = wh_gt[(b * KK + k) * 2 + 1];

  float num[SS];
  #pragma unroll
  for (int s = 0; s < SS; ++s) {
    size_t base = ((size_t)(s * BB + b)) * 2 * HWN;     // (S,B,2,H,W)
    float rp0 = reg_map[base + iv];
    float rp1 = reg_map[base + HWN + iv];
    float wp  = wh_map[base + iv];
    float hp  = wh_map[base + HWN + iv];
    float adx = fabsf(rg0 - rp0);
    float ady = fabsf(rg1 - rp1);
    float ldx = 1.0f - fmaxf((wg - 2.0f * adx) * hw_rcp(wg + 2.0f * adx + EPSF), 0.0f);
    float ldy = 1.0f - fmaxf((hg - 2.0f * ady) * hw_rcp(hg + 2.0f * ady + EPSF), 0.0f);
    float ldw = 1.0f - fminf(wg * hw_rcp(wp + EPSF), wp * hw_rcp(wg + EPSF));
    float ldh = 1.0f - fminf(hg * hw_rcp(hp + EPSF), hp * hw_rcp(hg + EPSF));
    num[s] = (smooth_l1(ldx) + smooth_l1(ldy) + smooth_l1(ldw) + smooth_l1(ldh)) * mf;
  }

  float n0 = wave_sum32(num[0]);
  float n1 = wave_sum32(num[1]);
  float ms = wave_sum32(mf);
  if (lane == 0) { sr[wave][0] = n0; sr[wave][1] = n1; sr[wave][2] = ms; }
  __syncthreads();
  if (k == 0) {
    float a0 = 0.f, a1 = 0.f, am = 0.f;
    for (int w = 0; w < 4; ++w) { a0 += sr[w][0]; a1 += sr[w][1]; am += sr[w][2]; }
    float den   = fmaxf(am, 1.0f) * 4.0f;
    float valid = (am > 0.0f) ? 1.0f : 0.0f;
    ws[32 + b]  = (a0 + a1) / den * valid / (float)(SS * BB);
  }
}

// ---------------------------------------------------------------------------
// Kernel 3: deterministic fixed-order final reduction.
// out = [loss, hm_loss, iou_loss]
// ---------------------------------------------------------------------------
__global__ void final_kernel(const float* __restrict__ ws, float* __restrict__ out) {
  if (threadIdx.x == 0 && blockIdx.x == 0) {
    float hm = 0.0f;
    for (int i = 0; i < 32; ++i) hm += ws[i];
    float io = 0.0f;
    for (int i = 0; i < 16; ++i) io += ws[32 + i];
    out[0] = 1.0f * hm + 0.1f * io;   // HM_WEIGHT, WH_WEIGHT
    out[1] = hm;
    out[2] = io;
  }
}

extern "C" void kernel_launch(void* const* d_in, const int* in_sizes, int n_in,
                              void* d_out, int out_size, void* d_ws, size_t ws_size,
                              hipStream_t stream) {
  const float* hm       = (const float*)d_in[0];
  const float* wh_map   = (const float*)d_in[1];
  const float* reg_map  = (const float*)d_in[2];
  const float* reg_gt   = (const float*)d_in[3];
  const float* wh_gt    = (const float*)d_in[4];
  const int*   ind      = (const int*)d_in[5];
  const int*   ctr      = (const int*)d_in[6];
  const int*   reg_mask = (const int*)d_in[7];
  float* ws  = (float*)d_ws;
  float* out = (float*)d_out;

  const size_t shmem = (size_t)(HWN + KK + 8) * sizeof(float);  // ~66 KB
  whd_kernel<<<SS * BB, 256, shmem, stream>>>(hm, ctr, reg_mask, ws);
  iou_kernel<<<BB, KK, 0, stream>>>(wh_map, reg_map, reg_gt, wh_gt, ind, reg_mask, ws);
  final_kernel<<<1, 32, 0, stream>>>(ws, out);
}